// PCA_Pointhop_57432302682839
// MI455X (gfx1250) — compile-verified
//
#include <hip/hip_runtime.h>
#include <math.h>

#define TPB     128            // threads per block = groups per block (4 waves)
#define K_PTS   32
#define OUT_F   39

#if defined(__HIP_DEVICE_COMPILE__) && __has_builtin(__builtin_amdgcn_tensor_load_to_lds)
#define USE_TDM 1
#else
#define USE_TDM 0
#endif

#if USE_TDM
#define SLOT_DW 99    // 96 data dwords + 3 TDM pad dwords (1 per 32) -> odd stride, bank-conflict-free
#else
#define SLOT_DW 100   // manual padding: 400 B slots, 16B-aligned async b128 destinations
#endif
#define CTR_F   (TPB * 100)    // float index of the centers slab (path-independent)

// map logical data dword d (within a group) to its LDS dword within the slot
__device__ __forceinline__ int didx(int d) {
#if USE_TDM
    return d + (d >> 5);       // TDM inserts 1 pad dword after every 32 data dwords
#else
    return d;                  // packed 96 dwords, tail-padded slot
#endif
}

#if USE_TDM
typedef unsigned int u32x4 __attribute__((ext_vector_type(4)));
typedef int          i32x4 __attribute__((ext_vector_type(4)));
typedef int          i32x8 __attribute__((ext_vector_type(8)));

// Issue one 2D TDM load: tile_rows x row_elems dwords, row-major, optional LDS padding.
// Rows >= tensor_rows are hardware-zero-filled (safe tail handling, no OOB access).
__device__ __forceinline__
void tdm_load_2d(unsigned lds_byte_addr, const void* gaddr,
                 unsigned row_elems, unsigned tensor_rows, unsigned tile_rows,
                 bool pad)
{
    unsigned long long ga = (unsigned long long)gaddr;
    u32x4 g0;
    g0.x = 1u;                                    // count=1 (valid), user mode
    g0.y = lds_byte_addr;                         // lds_addr [63:32]
    g0.z = (unsigned)(ga & 0xFFFFFFFFu);          // global_addr[31:0]
    g0.w = (unsigned)((ga >> 32) & 0x01FFFFFFu)   // global_addr[56:32]
         | 0x80000000u;                           // type = 2 ("image")

    unsigned d0 = (2u << 16);                     // data_size = 4 bytes; wg_mask=0
    if (pad) d0 |= (1u << 20)                     // pad_enable
               |  (4u << 22)                      // pad_interval code 4 = 32 dwords
               |  (0u << 25);                     // pad_amount  code 0 = 1 dword
    i32x8 g1;
    g1[0] = (int)d0;
    g1[1] = (int)(row_elems << 16);               // [47:32]=atomic addr 0 | tensor_dim0 lo16
    g1[2] = (int)((row_elems >> 16) | (tensor_rows << 16));              // dim0 hi | tensor_dim1 lo
    g1[3] = (int)(((tensor_rows >> 16) & 0xFFFFu) | (row_elems << 16));  // dim1 hi | tile_dim0
    g1[4] = (int)(tile_rows & 0xFFFFu);           // tile_dim1 | tile_dim2 = 0
    g1[5] = (int)row_elems;                       // tensor_dim0_stride[31:0]
    g1[6] = 0;                                    // stride hi | tensor_dim1_stride lo (2D: unused)
    g1[7] = 0;

    i32x4 z4 = {0, 0, 0, 0};
#if __has_include(<hip/amd_detail/amd_gfx1250_TDM.h>)
    i32x8 z8 = {0, 0, 0, 0, 0, 0, 0, 0};
    __builtin_amdgcn_tensor_load_to_lds(g0, g1, z4, z4, z8, 0);   // clang-23 / therock: 6-arg
#else
    __builtin_amdgcn_tensor_load_to_lds(g0, g1, z4, z4, 0);       // ROCm 7.2 clang-22: 5-arg
#endif
}
#endif // USE_TDM

// ---------- small helpers (all scalar, stays in VGPRs) ----------

// Mobius inversion: octant values from indicator moments.
// o = 4a + 2b + c with a=(x>0), b=(y>0), c=(z>0).
__device__ __forceinline__
void expand_octants(float S, float A, float B, float C,
                    float AB, float AC, float BC, float ABC, float o[8]) {
    o[7] = ABC;                         // a=1 b=1 c=1
    o[6] = AB - ABC;                    // a=1 b=1 c=0
    o[5] = AC - ABC;                    // a=1 b=0 c=1
    o[3] = BC - ABC;                    // a=0 b=1 c=1
    o[4] = A - AB - o[5];               // = A - AB - AC + ABC
    o[2] = B - AB - o[3];               // = B - AB - BC + ABC
    o[1] = C - AC - o[3];               // = C - AC - BC + ABC
    o[0] = S - A - B - C + AB + AC + BC - ABC;
}

__device__ __forceinline__ void sym3_eigvec(float a00, float a01, float a02,
                                            float a11, float a12, float a22,
                                            float lam,
                                            float& vx, float& vy, float& vz) {
    float r0x = a00 - lam, r0y = a01,       r0z = a02;
    float r1x = a01,       r1y = a11 - lam, r1z = a12;
    float r2x = a02,       r2y = a12,       r2z = a22 - lam;
    float c0x = r0y*r1z - r0z*r1y, c0y = r0z*r1x - r0x*r1z, c0z = r0x*r1y - r0y*r1x;
    float c1x = r0y*r2z - r0z*r2y, c1y = r0z*r2x - r0x*r2z, c1z = r0x*r2y - r0y*r2x;
    float c2x = r1y*r2z - r1z*r2y, c2y = r1z*r2x - r1x*r2z, c2z = r1x*r2y - r1y*r2x;
    float n0 = c0x*c0x + c0y*c0y + c0z*c0z;
    float n1 = c1x*c1x + c1y*c1y + c1z*c1z;
    float n2 = c2x*c2x + c2y*c2y + c2z*c2z;
    float nb = n0, bx = c0x, by = c0y, bz = c0z;
    if (n1 > nb) { nb = n1; bx = c1x; by = c1y; bz = c1z; }
    if (n2 > nb) { nb = n2; bx = c2x; by = c2y; bz = c2z; }
    if (nb < 1e-30f) { vx = 1.f; vy = 0.f; vz = 0.f; return; }
    float inv = rsqrtf(nb);
    vx = bx * inv; vy = by * inv; vz = bz * inv;
}

__device__ __forceinline__ void sign_fix(float& vx, float& vy, float& vz) {
    float ax = fabsf(vx), ay = fabsf(vy), az = fabsf(vz);
    float m;
    if (ax >= ay && ax >= az)      m = vx;
    else if (ay >= az)             m = vy;
    else                           m = vz;
    if (m < 0.f) { vx = -vx; vy = -vy; vz = -vz; }
}

// --------------------------------------------------------------------------

__global__ __launch_bounds__(TPB)
void pca_pointhop_kernel(const float* __restrict__ group_xyz,
                         const float* __restrict__ new_xyz,
                         float* __restrict__ out,
                         int n_groups)
{
    __shared__ float smem[TPB * 100 + TPB * 3];   // 52736 B (sized for both layouts)

    const int lane   = threadIdx.x & 31;
    const int wave   = threadIdx.x >> 5;
    const int gwave  = blockIdx.x * TPB + wave * 32;   // first group of this wave
    const int wslot  = wave * 32;                      // first LDS slot of this wave

    int gcount = n_groups - gwave;
    if (gcount > 32) gcount = 32;
    if (gcount < 0)  gcount = 0;

#if USE_TDM
    // ---- stage via Tensor Data Mover: one DMA per wave for the 32 groups
    //      (32 rows x 96 dwords, hw-padded to 99-dword slots), one for centers.
    {
        const unsigned lds_base = (unsigned)(unsigned long long)(const void*)&smem[0];
        tdm_load_2d(lds_base + (unsigned)(wslot * SLOT_DW * 4),
                    group_xyz + (size_t)gwave * (K_PTS * 3),
                    96u, (unsigned)gcount, 32u, /*pad=*/true);
        tdm_load_2d(lds_base + (unsigned)((CTR_F + wslot * 3) * 4),
                    new_xyz + (size_t)gwave * 3,
                    3u, (unsigned)gcount, 32u, /*pad=*/false);
        __builtin_amdgcn_s_wait_tensorcnt((short)0);   // both TDM ops (in-order per wave)
        asm volatile("" ::: "memory");
    }
#else
    // ---- fallback: per-lane async global->LDS b128 staging
    {
        const char* gsrc = (const char*)group_xyz + (size_t)gwave * (K_PTS * 3 * 4);
        const unsigned lds_base = (unsigned)(unsigned long long)(const void*)&smem[0];
        const int nchunk = gcount * 24;
        #pragma unroll
        for (int j = 0; j < 24; ++j) {
            int i = j * 32 + lane;
            if (i < nchunk) {
                int gl = i / 24;
                int c  = i - gl * 24;
                unsigned dst = lds_base + (unsigned)(((wslot + gl) * SLOT_DW) * 4 + c * 16);
                unsigned long long ga = (unsigned long long)(const void*)(gsrc + (size_t)i * 16u);
                asm volatile("global_load_async_to_lds_b128 %0, %1, off"
                             :: "v"(dst), "v"(ga) : "memory");
            }
        }
        asm volatile("s_wait_asynccnt 0" ::: "memory");
    }
#endif
    // Each wave only consumes its own staged region -> no workgroup barrier needed.

    if (lane < gcount) {
        const int g = gwave + lane;
        const float* __restrict__ my = &smem[(wslot + lane) * SLOT_DW];

        float sx = 0.f, sy = 0.f, sz = 0.f;
        float sxx = 0.f, syy = 0.f, szz = 0.f;
        float sxy = 0.f, sxz = 0.f, syz = 0.f;
        // indicator moments: counts and per-coordinate sums over a,b,c subsets
        float cA=0.f,cB=0.f,cC=0.f,cAB=0.f,cAC=0.f,cBC=0.f,cABC=0.f;
        float xA=0.f,xB=0.f,xC=0.f,xAB=0.f,xAC=0.f,xBC=0.f,xABC=0.f;
        float yA=0.f,yB=0.f,yC=0.f,yAB=0.f,yAC=0.f,yBC=0.f,yABC=0.f;
        float zA=0.f,zB=0.f,zC=0.f,zAB=0.f,zAC=0.f,zBC=0.f,zABC=0.f;

        #pragma unroll    // full unroll: didx() folds to DS immediate offsets
        for (int k = 0; k < K_PTS; ++k) {
            float x = my[didx(3*k + 0)];
            float y = my[didx(3*k + 1)];
            float z = my[didx(3*k + 2)];
            sx += x; sy += y; sz += z;
            sxx = fmaf(x, x, sxx); syy = fmaf(y, y, syy); szz = fmaf(z, z, szz);
            sxy = fmaf(x, y, sxy); sxz = fmaf(x, z, sxz); syz = fmaf(y, z, syz);

            float a = (x > 0.f) ? 1.f : 0.f;
            float b = (y > 0.f) ? 1.f : 0.f;
            float c = (z > 0.f) ? 1.f : 0.f;
            float ab = a*b, ac = a*c, bc = b*c, abc = ab*c;
            cA += a;  cB += b;  cC += c;  cAB += ab; cAC += ac; cBC += bc; cABC += abc;
            xA  = fmaf(a,  x, xA);  xB  = fmaf(b,  x, xB);  xC  = fmaf(c,  x, xC);
            xAB = fmaf(ab, x, xAB); xAC = fmaf(ac, x, xAC); xBC = fmaf(bc, x, xBC);
            xABC= fmaf(abc,x, xABC);
            yA  = fmaf(a,  y, yA);  yB  = fmaf(b,  y, yB);  yC  = fmaf(c,  y, yC);
            yAB = fmaf(ab, y, yAB); yAC = fmaf(ac, y, yAC); yBC = fmaf(bc, y, yBC);
            yABC= fmaf(abc,y, yABC);
            zA  = fmaf(a,  z, zA);  zB  = fmaf(b,  z, zB);  zC  = fmaf(c,  z, zC);
            zAB = fmaf(ab, z, zAB); zAC = fmaf(ac, z, zAC); zBC = fmaf(bc, z, zBC);
            zABC= fmaf(abc,z, zABC);
        }

        float res[OUT_F];

        // ---- std (ddof = 1)
        const float Kf = (float)K_PTS;
        float mx = sx / Kf, mmy = sy / Kf, mz = sz / Kf;
        float inv_km1 = 1.f / (Kf - 1.f);
        res[0] = sqrtf(fmaxf((sxx - Kf * mx  * mx ) * inv_km1, 0.f));
        res[1] = sqrtf(fmaxf((syy - Kf * mmy * mmy) * inv_km1, 0.f));
        res[2] = sqrtf(fmaxf((szz - Kf * mz  * mz ) * inv_km1, 0.f));

        // ---- center
#if USE_TDM
        res[3] = smem[CTR_F + (wslot + lane) * 3 + 0];
        res[4] = smem[CTR_F + (wslot + lane) * 3 + 1];
        res[5] = smem[CTR_F + (wslot + lane) * 3 + 2];
#else
        res[3] = new_xyz[3*g + 0];
        res[4] = new_xyz[3*g + 1];
        res[5] = new_xyz[3*g + 2];
#endif
        (void)g;

        // ---- octant means (8 x 3, e-major): Mobius inversion of the moments
        {
            float ocnt[8], osx[8], osy[8], osz[8];
            expand_octants(Kf, cA, cB, cC, cAB, cAC, cBC, cABC, ocnt);  // counts: exact
            expand_octants(sx, xA, xB, xC, xAB, xAC, xBC, xABC, osx);
            expand_octants(sy, yA, yB, yC, yAB, yAC, yBC, yABC, osy);
            expand_octants(sz, zA, zB, zC, zAB, zAC, zBC, zABC, osz);
            #pragma unroll
            for (int o = 0; o < 8; ++o) {
                float inv = 1.f / fmaxf(ocnt[o], 1.f);
                res[6 + 3*o + 0] = osx[o] * inv;
                res[6 + 3*o + 1] = osy[o] * inv;
                res[6 + 3*o + 2] = osz[o] * inv;
            }
        }

        // ---- eigensolve of uncentered Gram M = X^T X (closed form, 3x3 sym)
        float q  = (sxx + syy + szz) * (1.f / 3.f);
        float p1 = sxy*sxy + sxz*sxz + syz*syz;
        float b00 = sxx - q, b11 = syy - q, b22 = szz - q;
        float p2 = b00*b00 + b11*b11 + b22*b22 + 2.f * p1;
        float p  = sqrtf(p2 * (1.f / 6.f));
        float invp = (p > 1e-12f) ? (1.f / p) : 0.f;
        float c00 = b00*invp, c01 = sxy*invp, c02 = sxz*invp;
        float c11 = b11*invp, c12 = syz*invp, c22 = b22*invp;
        float detB = c00*(c11*c22 - c12*c12)
                   - c01*(c01*c22 - c12*c02)
                   + c02*(c01*c12 - c11*c02);
        float r = 0.5f * detB;
        r = fminf(1.f, fmaxf(-1.f, r));
        float phi = acosf(r) * (1.f / 3.f);
        float tp  = 2.f * p;
        float l1 = q + tp * cosf(phi);
        float l3 = q + tp * cosf(phi + 2.09439510239319549f);   // + 2*pi/3
        float l2 = 3.f * q - l1 - l3;                           // l1 >= l2 >= l3

        float s1 = sqrtf(fmaxf(l1, 0.f));
        float s2 = sqrtf(fmaxf(l2, 0.f));
        float s3 = sqrtf(fmaxf(l3, 0.f));

        float inva1 = 1.f / (s1 + 1e-10f);
        res[30] = (s1 - s2) * inva1;   // Linearity
        res[31] = (s2 - s3) * inva1;   // Planarity
        res[32] = s3 * inva1;          // Scattering

        float v1x, v1y, v1z, v2x, v2y, v2z, v3x, v3y, v3z;
        sym3_eigvec(sxx, sxy, sxz, syy, syz, szz, l1, v1x, v1y, v1z);
        sym3_eigvec(sxx, sxy, sxz, syy, syz, szz, l3, v3x, v3y, v3z);
        v2x = v3y*v1z - v3z*v1y;       // middle eigenvector = v3 x v1
        v2y = v3z*v1x - v3x*v1z;
        v2z = v3x*v1y - v3y*v1x;
        float n2 = v2x*v2x + v2y*v2y + v2z*v2z;
        if (n2 > 1e-30f) { float inv = rsqrtf(n2); v2x *= inv; v2y *= inv; v2z *= inv; }
        else             { v2x = 0.f; v2y = 1.f; v2z = 0.f; }
        sign_fix(v1x, v1y, v1z);
        sign_fix(v2x, v2y, v2z);
        sign_fix(v3x, v3y, v3z);

        // direction[j] = sum_i s_i * |vh[i][j]|   (sign-invariant)
        res[33] = s1*fabsf(v1x) + s2*fabsf(v2x) + s3*fabsf(v3x);
        res[34] = s1*fabsf(v1y) + s2*fabsf(v2y) + s3*fabsf(v3y);
        res[35] = s1*fabsf(v1z) + s2*fabsf(v2z) + s3*fabsf(v3z);

        // norm = vh[:, :, 0] = x-components of the three eigenvectors
        res[36] = v1x;
        res[37] = v2x;
        res[38] = v3x;

        // stage result back into OWN LDS slot (already fully consumed above)
        float* slot = &smem[(wslot + lane) * SLOT_DW];
        #pragma unroll
        for (int i = 0; i < OUT_F; ++i) slot[i] = res[i];
    }

    __syncthreads();

    // ---- coalesced, non-temporal store: lane-consecutive 128 B per wave,
    //      incremental (div-free) slot/column tracking.
    {
        const int outbase = gwave * OUT_F;
        const int nout    = gcount * OUT_F;
        int c      = lane;                         // lane < 32 < OUT_F -> gl = 0
        int ldsoff = wslot * SLOT_DW + lane;
        #pragma unroll
        for (int j = 0; j < OUT_F; ++j) {
            int idx = j * 32 + lane;
            if (idx < nout) {
                __builtin_nontemporal_store(smem[ldsoff], &out[outbase + idx]);
            }
            c += 32;
            int wrap = (c >= OUT_F) ? 1 : 0;
            c -= wrap * OUT_F;
            ldsoff += 32 + wrap * (SLOT_DW - OUT_F);
        }
    }
}

extern "C" void kernel_launch(void* const* d_in, const int* in_sizes, int n_in,
                              void* d_out, int out_size, void* d_ws, size_t ws_size,
                              hipStream_t stream) {
    const float* group_xyz = (const float*)d_in[0];
    const float* new_xyz   = (const float*)d_in[1];
    float* outp            = (float*)d_out;

    const int n_groups = in_sizes[1] / 3;              // B*N = 65536
    const int blocks   = (n_groups + TPB - 1) / TPB;   // 512

    pca_pointhop_kernel<<<dim3(blocks), dim3(TPB), 0, stream>>>(
        group_xyz, new_xyz, outp, n_groups);
}